// EV_GCN_80178449481840
// MI455X (gfx1250) — compile-verified
//
#include <hip/hip_runtime.h>
#include <hip/hip_bf16.h>

// ---------------------------------------------------------------------------
// EV-GCN inference for MI455X (gfx1250), fp32 WMMA path + CSR gather SpMM.
// N=20000 nodes, E=320000 edges, D=HGC=256, LG=4 cheb layers, NC=2 classes.
// ---------------------------------------------------------------------------

#define GN 20000
#define GE 320000

typedef float v2f __attribute__((ext_vector_type(2)));
typedef float v8f __attribute__((ext_vector_type(8)));

// ---------------------------------------------------------------------------
// Small utility kernels
// ---------------------------------------------------------------------------
__global__ void zero_f_kernel(float* __restrict__ p, int count) {
    int tid = blockIdx.x * blockDim.x + threadIdx.x;
    if (tid < count) p[tid] = 0.0f;
}

__global__ void zero_i_kernel(int* __restrict__ p, int count) {
    int tid = blockIdx.x * blockDim.x + threadIdx.x;
    if (tid < count) p[tid] = 0;
}

__global__ void copy_i_kernel(const int* __restrict__ src, int* __restrict__ dst, int count) {
    int tid = blockIdx.x * blockDim.x + threadIdx.x;
    if (tid < count) dst[tid] = src[tid];
}

__global__ void deg_scatter_kernel(const int* __restrict__ row,
                                   const float* __restrict__ ew,
                                   float* __restrict__ deg, int e_cnt) {
    int e = blockIdx.x * blockDim.x + threadIdx.x;
    if (e >= e_cnt) return;
    __hip_atomic_fetch_add(&deg[row[e]], ew[e],
                           __ATOMIC_RELAXED, __HIP_MEMORY_SCOPE_AGENT);
}

__global__ void dis_kernel(float* __restrict__ deg, int n_cnt) {
    int n = blockIdx.x * blockDim.x + threadIdx.x;
    if (n >= n_cnt) return;
    float d = deg[n];
    deg[n] = (d > 0.0f) ? __frsqrt_rn(d) : 0.0f;  // deg buffer becomes dis
}

__global__ void norm_kernel(const int* __restrict__ row, const int* __restrict__ col,
                            const float* __restrict__ ew, const float* __restrict__ dis,
                            float* __restrict__ norm, float* __restrict__ out_ew,
                            int e_cnt) {
    int e = blockIdx.x * blockDim.x + threadIdx.x;
    if (e >= e_cnt) return;
    float w = ew[e];
    norm[e] = -(dis[row[e]] * w * dis[col[e]]);
    out_ew[e] = w;  // second output of the reference
}

// ---------------------------------------------------------------------------
// CSR build (bucket edges by target node `col`)
// ---------------------------------------------------------------------------
__global__ void count_kernel(const int* __restrict__ col, int* __restrict__ counts, int e_cnt) {
    int e = blockIdx.x * blockDim.x + threadIdx.x;
    if (e >= e_cnt) return;
    atomicAdd(&counts[col[e]], 1);
}

// Single-workgroup exclusive scan over n elements (n ~ 20000: ~80 chunks).
__global__ void scan_kernel(const int* __restrict__ counts, int* __restrict__ rowptr, int n) {
    __shared__ int smem[256];
    __shared__ int carry_s;
    int tid = threadIdx.x;
    if (tid == 0) carry_s = 0;
    __syncthreads();
    for (int base = 0; base < n; base += 256) {
        int v = (base + tid < n) ? counts[base + tid] : 0;
        smem[tid] = v;
        __syncthreads();
#pragma unroll
        for (int off = 1; off < 256; off <<= 1) {
            int t = (tid >= off) ? smem[tid - off] : 0;
            __syncthreads();
            smem[tid] += t;
            __syncthreads();
        }
        int carry = carry_s;
        if (base + tid < n) rowptr[base + tid] = carry + smem[tid] - v;  // exclusive
        __syncthreads();
        if (tid == 255) carry_s = carry + smem[255];
        __syncthreads();
    }
    if (tid == 0) rowptr[n] = carry_s;
}

__global__ void csr_fill_kernel(const int* __restrict__ col, int* __restrict__ cursor,
                                int* __restrict__ eidx, int e_cnt) {
    int e = blockIdx.x * blockDim.x + threadIdx.x;
    if (e >= e_cnt) return;
    int pos = atomicAdd(&cursor[col[e]], 1);
    eidx[pos] = e;
}

// ---------------------------------------------------------------------------
// Gather-form propagation: out[n,:] = scale * sum_{e in CSR[n]} norm[e]*x[row[e],:]
//                                     - (sub ? sub[n,:] : 0)
// One wave per node; full 256-float row accumulated in registers (8/lane),
// written once -> zero float atomics. x-row reads are two coalesced b128
// bursts per edge; index/norm reads are wave-uniform (scalar broadcast).
// out has stride 256. With sub!=null this computes the Chebyshev T2 term
// 2*prop(tx1) - x directly.
// ---------------------------------------------------------------------------
__global__ void prop_gather_kernel(const float* __restrict__ x, int ldx,
                                   const int* __restrict__ rowptr,
                                   const int* __restrict__ eidx,
                                   const int* __restrict__ row,
                                   const float* __restrict__ norm,
                                   float scale,
                                   const float* __restrict__ sub, int ldsub,
                                   float* __restrict__ out, int n_cnt) {
    int node = (blockIdx.x * blockDim.x + threadIdx.x) >> 5;
    int lane = threadIdx.x & 31;
    if (node >= n_cnt) return;
    int s = rowptr[node];
    int t = rowptr[node + 1];
    float4 acc0 = make_float4(0.f, 0.f, 0.f, 0.f);
    float4 acc1 = make_float4(0.f, 0.f, 0.f, 0.f);
    int f0 = lane * 4;        // features [0,128)
    int f1 = 128 + lane * 4;  // features [128,256)
    for (int i = s; i < t; ++i) {
        int e = eidx[i];
        float w = norm[e];
        const float* xr = x + (size_t)row[e] * ldx;
        float4 a = *(const float4*)(xr + f0);
        float4 b = *(const float4*)(xr + f1);
        acc0.x = fmaf(w, a.x, acc0.x); acc0.y = fmaf(w, a.y, acc0.y);
        acc0.z = fmaf(w, a.z, acc0.z); acc0.w = fmaf(w, a.w, acc0.w);
        acc1.x = fmaf(w, b.x, acc1.x); acc1.y = fmaf(w, b.y, acc1.y);
        acc1.z = fmaf(w, b.z, acc1.z); acc1.w = fmaf(w, b.w, acc1.w);
    }
    float4 r0, r1;
    if (sub) {
        const float* sr = sub + (size_t)node * ldsub;
        float4 s0 = *(const float4*)(sr + f0);
        float4 s1 = *(const float4*)(sr + f1);
        r0 = make_float4(scale * acc0.x - s0.x, scale * acc0.y - s0.y,
                         scale * acc0.z - s0.z, scale * acc0.w - s0.w);
        r1 = make_float4(scale * acc1.x - s1.x, scale * acc1.y - s1.y,
                         scale * acc1.z - s1.z, scale * acc1.w - s1.w);
    } else {
        r0 = acc0;
        r1 = acc1;
    }
    float* o = out + (size_t)node * 256;
    *(float4*)(o + f0) = r0;
    *(float4*)(o + f1) = r1;
}

// ---------------------------------------------------------------------------
// Fused multi-pair WMMA GEMM:  out = epilogue( sum_p A_p @ B_p )
//   A_p: [M, K] row-major (stride lda_p), B_p: [K, 256] row-major (stride 256)
//   out: [M, 256] written at column stride ldout.
// One wave computes a 16x64 output strip (4 accumulators, A-frag reuse x4).
// Uses V_WMMA_F32_16X16X4_F32 (exact fp32 matrix pipe on CDNA5).
// M must be a multiple of 16 (20000 = 1250*16) -> EXEC all-ones everywhere.
// ---------------------------------------------------------------------------
__device__ __forceinline__ void wmma_accum_pair(const float* __restrict__ A, int lda,
                                                const float* __restrict__ B,
                                                int rowbase, int colbase, int K,
                                                int l15, int half, v8f c[4]) {
    // A fragment layout (16x4 f32): lane<16 holds M=lane, K={k,k+1};
    //                               lane>=16 holds M=lane-16, K={k+2,k+3}.
    const float* arow = A + (size_t)(rowbase + l15) * lda + half * 2;
    const float* bcol = B + colbase + l15 + (size_t)(half * 2) * 256;
    for (int k = 0; k < K; k += 4) {
        v2f a;
        a.x = arow[k];
        a.y = arow[k + 1];
        const float* brow = bcol + (size_t)k * 256;
#pragma unroll
        for (int t = 0; t < 4; ++t) {
            v2f b;
            b.x = brow[t * 16];
            b.y = brow[t * 16 + 256];
            c[t] = __builtin_amdgcn_wmma_f32_16x16x4_f32(
                false, a, false, b, (short)0, c[t], false, false);
        }
    }
}

__global__ void gemm_wmma_kernel(const float* __restrict__ A0, int lda0, const float* __restrict__ B0,
                                 const float* __restrict__ A1, int lda1, const float* __restrict__ B1,
                                 const float* __restrict__ A2, int lda2, const float* __restrict__ B2,
                                 int K, float* __restrict__ out, int ldout,
                                 const float* __restrict__ bias,
                                 const float* __restrict__ bng, const float* __restrict__ bnb,
                                 const float* __restrict__ bnm, const float* __restrict__ bnv) {
    int wave = (blockIdx.x * blockDim.x + threadIdx.x) >> 5;  // 8 waves / block
    int lane = threadIdx.x & 31;
    int l15 = lane & 15;
    int half = lane >> 4;

    int rowTile = wave >> 2;        // 1250 row tiles of 16
    int colStrip = wave & 3;        // 4 strips of 64 columns
    int rowbase = rowTile << 4;
    int colbase = colStrip << 6;

    v8f c[4];
#pragma unroll
    for (int t = 0; t < 4; ++t) c[t] = (v8f)(0.0f);

    wmma_accum_pair(A0, lda0, B0, rowbase, colbase, K, l15, half, c);
    if (A1) wmma_accum_pair(A1, lda1, B1, rowbase, colbase, K, l15, half, c);
    if (A2) wmma_accum_pair(A2, lda2, B2, rowbase, colbase, K, l15, half, c);

    // C/D layout: VGPR r holds M = r + 8*half, N = colbase + t*16 + l15.
#pragma unroll
    for (int t = 0; t < 4; ++t) {
        int n = colbase + t * 16 + l15;
        float add = bias ? bias[n] : 0.0f;
        float sc = 1.0f, mu = 0.0f, be = 0.0f;
        if (bng) {
            sc = bng[n] * __frsqrt_rn(bnv[n] + 1e-5f);
            mu = bnm[n];
            be = bnb[n];
        }
#pragma unroll
        for (int r = 0; r < 8; ++r) {
            int m = rowbase + r + 8 * half;
            float v = c[t][r] + add;
            v = fmaxf(v, 0.0f);                 // ReLU
            if (bng) v = (v - mu) * sc + be;    // BatchNorm (eval)
            out[(size_t)m * ldout + n] = v;
        }
    }
}

// logits = z @ w2 + b2, z:[M,256], w2:[256,2]. Wave-per-node shuffle reduce.
__global__ void logit_kernel(const float* __restrict__ z, const float* __restrict__ w2,
                             const float* __restrict__ b2, float* __restrict__ out, int M) {
    int wave = (blockIdx.x * blockDim.x + threadIdx.x) >> 5;
    int lane = threadIdx.x & 31;
    if (wave >= M) return;
    const float* zr = z + (size_t)wave * 256;
    float a0 = 0.0f, a1 = 0.0f;
#pragma unroll
    for (int i = 0; i < 8; ++i) {
        int k = lane + i * 32;
        float zv = zr[k];
        a0 += zv * w2[k * 2 + 0];
        a1 += zv * w2[k * 2 + 1];
    }
#pragma unroll
    for (int off = 16; off > 0; off >>= 1) {
        a0 += __shfl_xor(a0, off, 32);
        a1 += __shfl_xor(a1, off, 32);
    }
    if (lane == 0) {
        out[wave * 2 + 0] = a0 + b2[0];
        out[wave * 2 + 1] = a1 + b2[1];
    }
}

// ---------------------------------------------------------------------------
// Host-side orchestration
// ---------------------------------------------------------------------------
extern "C" void kernel_launch(void* const* d_in, const int* in_sizes, int n_in,
                              void* d_out, int out_size, void* d_ws, size_t ws_size,
                              hipStream_t stream) {
    const float* features = (const float*)d_in[0];      // [N,256]
    const int*   edge_index = (const int*)d_in[1];      // [2,E]
    // d_in[2] = edgenet_input (unused: override bypasses PAE)
    const float* ew   = (const float*)d_in[3];          // [E]
    const float* chw  = (const float*)d_in[4];          // [4,3,256,256]
    const float* w1   = (const float*)d_in[5];          // [1024,256]
    const float* b1   = (const float*)d_in[6];
    const float* bng  = (const float*)d_in[7];
    const float* bnb  = (const float*)d_in[8];
    const float* bnm  = (const float*)d_in[9];
    const float* bnv  = (const float*)d_in[10];
    const float* w2   = (const float*)d_in[11];         // [256,2]
    const float* b2   = (const float*)d_in[12];

    const int* row = edge_index;        // source
    const int* col = edge_index + GE;   // target

    float* out_logit = (float*)d_out;        // [N,2]
    float* out_ew    = out_logit + GN * 2;   // [E]

    // Workspace layout
    float* dis  = (float*)d_ws;                       // N      (deg -> dis)
    float* norm = dis + GN;                           // E
    float* jk   = norm + GE;                          // N*1024 (concat of 4 layers)
    float* tx1  = jk + (size_t)GN * 1024;             // N*256
    float* tx2  = tx1 + (size_t)GN * 256;             // N*256
    float* zb   = tx2 + (size_t)GN * 256;             // N*256
    int* rowptr = (int*)(zb + (size_t)GN * 256);      // N+1
    int* cursor = rowptr + (GN + 1);                  // N  (counts, then fill cursor)
    int* eidx   = cursor + GN;                        // E

    const int TB = 256;
    const int nN_b   = (GN + TB - 1) / TB;
    const int nE_b   = (GE + TB - 1) / TB;
    const int gemm_b = (1250 * 4) / 8;     // 625 blocks = 5000 waves
    const int wavN_b = (GN * 32) / TB;     // 2500 blocks: wave-per-node kernels

    // 1) normalized Laplacian edge weights
    zero_f_kernel<<<nN_b, TB, 0, stream>>>(dis, GN);
    deg_scatter_kernel<<<nE_b, TB, 0, stream>>>(row, ew, dis, GE);
    dis_kernel<<<nN_b, TB, 0, stream>>>(dis, GN);
    norm_kernel<<<nE_b, TB, 0, stream>>>(row, col, ew, dis, norm, out_ew, GE);

    // 2) CSR by target node (once per launch; removes all float atomics below)
    zero_i_kernel<<<nN_b, TB, 0, stream>>>(cursor, GN);
    count_kernel<<<nE_b, TB, 0, stream>>>(col, cursor, GE);
    scan_kernel<<<1, TB, 0, stream>>>(cursor, rowptr, GN);
    copy_i_kernel<<<nN_b, TB, 0, stream>>>(rowptr, cursor, GN);
    csr_fill_kernel<<<nE_b, TB, 0, stream>>>(col, cursor, eidx, GE);

    // 3) four ChebConv(K=3) layers, output concatenated into jk
    for (int l = 0; l < 4; ++l) {
        const float* x  = (l == 0) ? features : (jk + (size_t)(l - 1) * 256);
        int ldx         = (l == 0) ? 256 : 1024;
        const float* W  = chw + (size_t)l * 3 * 256 * 256;

        // tx1 = prop(x)
        prop_gather_kernel<<<wavN_b, TB, 0, stream>>>(
            x, ldx, rowptr, eidx, row, norm, 1.0f, nullptr, 0, tx1, GN);
        // tx2 = 2*prop(tx1) - x   (fused Chebyshev T2)
        prop_gather_kernel<<<wavN_b, TB, 0, stream>>>(
            tx1, 256, rowptr, eidx, row, norm, 2.0f, x, ldx, tx2, GN);

        gemm_wmma_kernel<<<gemm_b, TB, 0, stream>>>(
            x, ldx, W,                       // T0 @ W0
            tx1, 256, W + 256 * 256,         // T1 @ W1
            tx2, 256, W + 2 * 256 * 256,     // T2 @ W2
            256, jk + (size_t)l * 256, 1024,
            nullptr, nullptr, nullptr, nullptr, nullptr);
    }

    // 4) classifier hidden: relu(jk @ w1 + b1) then BatchNorm (fused epilogue)
    gemm_wmma_kernel<<<gemm_b, TB, 0, stream>>>(
        jk, 1024, w1,
        nullptr, 0, nullptr,
        nullptr, 0, nullptr,
        1024, zb, 256,
        b1, bng, bnb, bnm, bnv);

    // 5) final projection [256 -> 2]
    logit_kernel<<<wavN_b, TB, 0, stream>>>(zb, w2, b2, out_logit, GN);
}